// PNANet_45097156608287
// MI455X (gfx1250) — compile-verified
//
#include <hip/hip_runtime.h>
#include <hip/hip_bf16.h>

#define NN 50000
#define NE 800000
#define NG 500

typedef __attribute__((ext_vector_type(2))) float v2f;
typedef __attribute__((ext_vector_type(8))) float v8f;

// V_WMMA_F32_16X16X4_F32: D = A(16x4 f32) x B(4x16 f32) + C(16x16 f32)
// A layout (ISA 7.12.2): lane L, row M = L%16; VGPR0/1 hold K = {0,1} (L<16) or {2,3} (L>=16)
// B layout mirrors A with N across lanes; C/D: VGPR j holds M = j (+8 for L>=16), N = L%16.
__device__ __forceinline__ v8f wmma4(v2f a, v2f b, v8f c) {
  return __builtin_amdgcn_wmma_f32_16x16x4_f32(false, a, false, b, (short)0, c, false, false);
}

// LDS weight packing: for each 4-row K-block, the 4 values of a column are contiguous:
//   Wl[((k>>2)<<8) + (col<<2) + (k&3)] = W[k][col]
// so a lane's B fragment {W[k0+2h][col], W[k0+2h+1][col]} is one aligned 8-byte load.
__device__ __forceinline__ v2f bfrag(const float* Wl, int k0, int col, int half) {
  return *(const v2f*)&Wl[((k0 >> 2) << 8) + (col << 2) + 2 * half];
}
__device__ __forceinline__ v2f afrag(const float* ar, int kk) {
  return *(const v2f*)(ar + kk);  // kk even, row bases >=64B aligned
}

// order-preserving float <-> uint encoding for atomic max/min
__device__ __forceinline__ unsigned enc_ord(float x) {
  unsigned u = __float_as_uint(x);
  return (u & 0x80000000u) ? ~u : (u | 0x80000000u);
}
__device__ __forceinline__ float dec_ord(unsigned u) {
  unsigned b = (u & 0x80000000u) ? (u & 0x7FFFFFFFu) : ~u;
  return __uint_as_float(b);
}

// ---------------- degree (in-degree over dst) ----------------
__global__ __launch_bounds__(256)
void deg_kernel(const int* __restrict__ dst, float* __restrict__ deg, int nE) {
  int i = blockIdx.x * blockDim.x + threadIdx.x;
  if (i < nE) atomicAdd(&deg[dst[i]], 1.0f);
}

// ---------------- fold edge-embedding into message weights ----------------
// wfold[16,64] = We[16,64] @ Mw[64:128,64];  bfold = be @ Mw[64:128,:] + Mb
__global__ __launch_bounds__(256)
void fold_kernel(const float* __restrict__ We, const float* __restrict__ be,
                 const float* __restrict__ Mw, const float* __restrict__ Mb,
                 float* __restrict__ wfold, float* __restrict__ bfold) {
  int i = blockIdx.x * blockDim.x + threadIdx.x;
  if (i < 16 * 64) {
    int r = i >> 6, c = i & 63;
    float s = 0.f;
    for (int k = 0; k < 64; k++) s += We[r * 64 + k] * Mw[(64 + k) * 64 + c];
    wfold[i] = s;
  } else if (i < 17 * 64) {
    int c = i & 63;
    float s = Mb[c];
    for (int k = 0; k < 64; k++) s += be[k] * Mw[(64 + k) * 64 + c];
    bfold[c] = s;
  }
}

// ---------------- node embedding: h = h_in[NN,32] @ W[32,64] + b ----------------
__global__ __launch_bounds__(128)
void embed_kernel(const float* __restrict__ X, const float* __restrict__ W,
                  const float* __restrict__ bias, float* __restrict__ out, int n) {
  __shared__ __align__(16) float Wl[32 * 64];
  __shared__ float Bl[64];
  for (int i = threadIdx.x; i < 32 * 64; i += blockDim.x) {
    int k = i >> 6, col = i & 63;
    Wl[((k >> 2) << 8) + (col << 2) + (k & 3)] = W[i];
  }
  if (threadIdx.x < 64) Bl[threadIdx.x] = bias[threadIdx.x];
  __syncthreads();
  int wave = threadIdx.x >> 5, lane = threadIdx.x & 31;
  int half = lane >> 4, lm = lane & 15;
  int n0 = (blockIdx.x * 4 + wave) * 16;
  if (n0 >= n) return;
  int row = min(n0 + lm, n - 1);
  const float* xr = &X[(size_t)row * 32];
  v8f acc[4];
  #pragma unroll
  for (int t = 0; t < 4; t++) {
    float bv = Bl[t * 16 + lm];
    #pragma unroll
    for (int j = 0; j < 8; j++) acc[t][j] = bv;
  }
  for (int k0 = 0; k0 < 32; k0 += 4) {
    v2f a = afrag(xr, k0 + 2 * half);
    #pragma unroll
    for (int t = 0; t < 4; t++)
      acc[t] = wmma4(a, bfrag(Wl, k0, t * 16 + lm, half), acc[t]);
  }
  #pragma unroll
  for (int j = 0; j < 8; j++) {
    int m = half * 8 + j, nd = n0 + m;
    if (nd < n) {
      #pragma unroll
      for (int t = 0; t < 4; t++) out[(size_t)nd * 64 + t * 16 + lm] = acc[t][j];
    }
  }
}

// ---------------- fused message GEMM + segment aggregation ----------------
// msg = h[src]@Mw[0:64] + e_in@wfold + h[dst]@Mw[128:192] + bfold, scattered by dst.
__global__ __launch_bounds__(128)
void msg_agg_kernel(const float* __restrict__ h, const float* __restrict__ e_in,
                    const int* __restrict__ src, const int* __restrict__ dst,
                    const float* __restrict__ Mw, const float* __restrict__ wfold,
                    const float* __restrict__ bfold,
                    float* __restrict__ sum, float* __restrict__ sumsq,
                    unsigned* __restrict__ maxe, unsigned* __restrict__ mine,
                    int nE, int need_minstd) {
  // packed segments: [0..4095]=W_src(64 rows), [4096..5119]=wfold(16), [5120..9215]=W_dst(64)
  __shared__ __align__(16) float Wl[144 * 64];
  __shared__ float Bl[64];
  for (int i = threadIdx.x; i < 64 * 64; i += blockDim.x) {
    int k = i >> 6, col = i & 63;
    int pi = ((k >> 2) << 8) + (col << 2) + (k & 3);
    Wl[pi] = Mw[i];
    Wl[5120 + pi] = Mw[128 * 64 + i];
  }
  for (int i = threadIdx.x; i < 16 * 64; i += blockDim.x) {
    int k = i >> 6, col = i & 63;
    Wl[4096 + ((k >> 2) << 8) + (col << 2) + (k & 3)] = wfold[i];
  }
  if (threadIdx.x < 64) Bl[threadIdx.x] = bfold[threadIdx.x];
  __syncthreads();

  int wave = threadIdx.x >> 5, lane = threadIdx.x & 31;
  int half = lane >> 4, lm = lane & 15;
  int e0 = (blockIdx.x * 4 + wave) * 16;
  if (e0 >= nE) return;
  int me = e0 + lm;
  int erow = (me < nE) ? me : (nE - 1);
  int ns = src[erow];
  int nd = dst[erow];

  v8f acc[4];
  #pragma unroll
  for (int t = 0; t < 4; t++) {
    float bv = Bl[t * 16 + lm];
    #pragma unroll
    for (int j = 0; j < 8; j++) acc[t][j] = bv;
  }

  const float* a0 = &h[(size_t)ns * 64];
  const float* a1 = &e_in[(size_t)erow * 16];
  const float* a2 = &h[(size_t)nd * 64];

  auto seg = [&](const float* ar, int K, int woff) {
    for (int k0 = 0; k0 < K; k0 += 4) {
      v2f a = afrag(ar, k0 + 2 * half);
      #pragma unroll
      for (int t = 0; t < 4; t++)
        acc[t] = wmma4(a, bfrag(Wl + woff, k0, t * 16 + lm, half), acc[t]);
    }
  };
  seg(a0, 64, 0);
  seg(a1, 16, 4096);
  seg(a2, 64, 5120);

  #pragma unroll
  for (int j = 0; j < 8; j++) {
    int m = half * 8 + j;
    int edge = e0 + m;
    if (edge < nE) {
      int node = dst[edge];
      #pragma unroll
      for (int t = 0; t < 4; t++) {
        float v = acc[t][j];
        size_t idx = (size_t)node * 64 + t * 16 + lm;
        atomicAdd(&sum[idx], v);
        atomicMax(&maxe[idx], enc_ord(v));
        if (need_minstd) {
          atomicAdd(&sumsq[idx], v * v);
          atomicMin(&mine[idx], enc_ord(v));
        }
      }
    }
  }
}

// aggregator value for combo (agg id: 0=mean 1=max 2=min 3=std 4=sum)
__device__ __forceinline__ float agg_val(int agg, size_t idx,
                                         const float* __restrict__ sum,
                                         const float* __restrict__ sumsq,
                                         const unsigned* __restrict__ maxe,
                                         const unsigned* __restrict__ mine,
                                         float deg1, bool has) {
  switch (agg) {
    case 0: return sum[idx] / deg1;
    case 1: return has ? dec_ord(maxe[idx]) : 0.f;
    case 2: return has ? dec_ord(mine[idx]) : 0.f;
    case 3: {
      float mean = sum[idx] / deg1;
      float m2 = sumsq[idx] / deg1;
      return sqrtf(fmaxf(m2 - mean * mean, 0.f) + 1e-5f);
    }
    default: return sum[idx];
  }
}

// ---------------- node update GEMM: z = cat(h, h_agg) @ Uw + Ub ----------------
// A-fragments built on the fly from aggregation buffers; Uw staged in 16KB LDS chunks.
__global__ __launch_bounds__(128)
void u_kernel(const float* __restrict__ h, const float* __restrict__ sum,
              const float* __restrict__ sumsq, const unsigned* __restrict__ maxe,
              const unsigned* __restrict__ mine, const float* __restrict__ deg,
              const float* __restrict__ Uw, const float* __restrict__ Ub,
              float* __restrict__ zout, int n, int ltype) {
  __shared__ __align__(16) float Wl[64 * 64];
  int wave = threadIdx.x >> 5, lane = threadIdx.x & 31;
  int half = lane >> 4, lm = lane & 15;
  int n0 = (blockIdx.x * 4 + wave) * 16;
  bool active = n0 < n;
  int node = active ? min(n0 + lm, n - 1) : 0;
  float dg = deg[node];
  float deg1 = fmaxf(dg, 1.f);
  bool has = dg > 0.f;
  float amp = logf(dg + 1.f) * (1.f / 2.5f);
  float att = 2.5f / logf(fmaxf(dg, 1.f) + 1.f);
  int C = (ltype == 0) ? 12 : 6;
  const float* hrow = &h[(size_t)node * 64];
  size_t nb = (size_t)node * 64;

  v8f acc[4];
  #pragma unroll
  for (int t = 0; t < 4; t++) {
    float bv = Ub[t * 16 + lm];
    #pragma unroll
    for (int j = 0; j < 8; j++) acc[t][j] = bv;
  }

  for (int c = 0; c <= C; c++) {
    __syncthreads();
    for (int i = threadIdx.x; i < 64 * 64; i += blockDim.x) {
      int k = i >> 6, col = i & 63;
      Wl[((k >> 2) << 8) + (col << 2) + (k & 3)] = Uw[(size_t)c * 4096 + i];
    }
    __syncthreads();
    int agg = -1; float smul = 1.f;
    if (c > 0) {
      int cc = c - 1, scl;
      if (ltype == 0) { agg = cc / 3; scl = cc % 3; }                     // [mean,max,min,std] x [id,amp,att]
      else { int ai = cc >> 1; agg = (ai == 2) ? 4 : ai; scl = cc & 1; }  // [mean,max,sum] x [id,amp]
      smul = (scl == 0) ? 1.f : ((scl == 1) ? amp : att);
    }
    for (int k0 = 0; k0 < 64; k0 += 4) {
      int kk = k0 + 2 * half;
      v2f a;
      if (c == 0) { a = afrag(hrow, kk); }
      else {
        a.x = agg_val(agg, nb + kk, sum, sumsq, maxe, mine, deg1, has) * smul;
        a.y = agg_val(agg, nb + kk + 1, sum, sumsq, maxe, mine, deg1, has) * smul;
      }
      #pragma unroll
      for (int t = 0; t < 4; t++)
        acc[t] = wmma4(a, bfrag(Wl, k0, t * 16 + lm, half), acc[t]);
    }
  }
  if (active) {
    #pragma unroll
    for (int j = 0; j < 8; j++) {
      int m = half * 8 + j, nd = n0 + m;
      if (nd < n) {
        #pragma unroll
        for (int t = 0; t < 4; t++) zout[(size_t)nd * 64 + t * 16 + lm] = acc[t][j];
      }
    }
  }
}

// ---------------- batchnorm statistics over nodes ----------------
__global__ __launch_bounds__(256)
void bn_stats_kernel(const float* __restrict__ z, float* __restrict__ bnsum,
                     float* __restrict__ bnss, int n) {
  __shared__ float ls[256], lss[256];
  int f = threadIdx.x & 63, g = threadIdx.x >> 6;
  float s = 0.f, ss = 0.f;
  for (int i = blockIdx.x * 4 + g; i < n; i += gridDim.x * 4) {
    float v = z[(size_t)i * 64 + f];
    s += v; ss += v * v;
  }
  ls[threadIdx.x] = s; lss[threadIdx.x] = ss;
  __syncthreads();
  if (threadIdx.x < 64) {
    float t = ls[threadIdx.x] + ls[threadIdx.x + 64] + ls[threadIdx.x + 128] + ls[threadIdx.x + 192];
    float tt = lss[threadIdx.x] + lss[threadIdx.x + 64] + lss[threadIdx.x + 128] + lss[threadIdx.x + 192];
    atomicAdd(&bnsum[threadIdx.x], t);
    atomicAdd(&bnss[threadIdx.x], tt);
  }
}

__global__ __launch_bounds__(64)
void bn_affine_kernel(const float* __restrict__ bnsum, const float* __restrict__ bnss,
                      const float* __restrict__ g, const float* __restrict__ b,
                      float* __restrict__ alpha, float* __restrict__ beta, float n) {
  int f = threadIdx.x;
  if (f < 64) {
    float mu = bnsum[f] / n;
    float var = fmaxf(bnss[f] / n - mu * mu, 0.f);
    float a = g[f] * rsqrtf(var + 1e-5f);
    alpha[f] = a;
    beta[f] = b[f] - mu * a;
  }
}

// ---------------- BN + mix GEMM + leaky_relu + residual ----------------
__global__ __launch_bounds__(128)
void mix_kernel(const float* __restrict__ z, const float* __restrict__ alpha,
                const float* __restrict__ beta, const float* __restrict__ Wm,
                const float* __restrict__ bm, const float* __restrict__ hin,
                float* __restrict__ hout, int n) {
  __shared__ __align__(16) float Wl[64 * 64];
  __shared__ float Al[64], Be[64], Bm[64];
  for (int i = threadIdx.x; i < 64 * 64; i += blockDim.x) {
    int k = i >> 6, col = i & 63;
    Wl[((k >> 2) << 8) + (col << 2) + (k & 3)] = Wm[i];
  }
  if (threadIdx.x < 64) {
    Al[threadIdx.x] = alpha[threadIdx.x];
    Be[threadIdx.x] = beta[threadIdx.x];
    Bm[threadIdx.x] = bm[threadIdx.x];
  }
  __syncthreads();
  int wave = threadIdx.x >> 5, lane = threadIdx.x & 31;
  int half = lane >> 4, lm = lane & 15;
  int n0 = (blockIdx.x * 4 + wave) * 16;
  if (n0 >= n) return;
  int node = min(n0 + lm, n - 1);
  const float* zrow = &z[(size_t)node * 64];
  v8f acc[4];
  #pragma unroll
  for (int t = 0; t < 4; t++) {
    float bv = Bm[t * 16 + lm];
    #pragma unroll
    for (int j = 0; j < 8; j++) acc[t][j] = bv;
  }
  for (int k0 = 0; k0 < 64; k0 += 4) {
    int kk = k0 + 2 * half;
    v2f zv = afrag(zrow, kk);
    v2f a;
    a.x = zv.x * Al[kk] + Be[kk];
    a.y = zv.y * Al[kk + 1] + Be[kk + 1];
    #pragma unroll
    for (int t = 0; t < 4; t++)
      acc[t] = wmma4(a, bfrag(Wl, k0, t * 16 + lm, half), acc[t]);
  }
  #pragma unroll
  for (int j = 0; j < 8; j++) {
    int m = half * 8 + j, nd = n0 + m;
    if (nd < n) {
      #pragma unroll
      for (int t = 0; t < 4; t++) {
        float v = acc[t][j];
        v = (v > 0.f) ? v : 0.01f * v;  // leaky_relu
        size_t idx = (size_t)nd * 64 + t * 16 + lm;
        hout[idx] = v + hin[idx];       // residual
      }
    }
  }
}

// ---------------- graph mean pooling ----------------
__global__ __launch_bounds__(256)
void pool_kernel(const float* __restrict__ h, const int* __restrict__ gid,
                 float* __restrict__ hg, float* __restrict__ cnt, int n) {
  int i = blockIdx.x * blockDim.x + threadIdx.x;
  if (i >= n * 64) return;
  int node = i >> 6, f = i & 63;
  int g = gid[node];
  atomicAdd(&hg[(size_t)g * 64 + f], h[i]);
  if (f == 0) atomicAdd(&cnt[g], 1.f);
}

// ---------------- final MLP: 64 -> 32 -> 16 -> 2 ----------------
__global__ __launch_bounds__(64)
void mlp_kernel(const float* __restrict__ hg, const float* __restrict__ cnt,
                const float* __restrict__ b1, const float* __restrict__ w1,
                const float* __restrict__ b2, const float* __restrict__ w2,
                const float* __restrict__ b3, const float* __restrict__ w3,
                float* __restrict__ out) {
  __shared__ float x[64], y1[32], y2[16];
  int g = blockIdx.x, t = threadIdx.x;
  float c = fmaxf(cnt[g], 1.f);
  x[t] = hg[(size_t)g * 64 + t] / c;
  __syncthreads();
  if (t < 32) { float s = b1[t]; for (int k = 0; k < 64; k++) s += x[k] * w1[k * 32 + t]; y1[t] = fmaxf(s, 0.f); }
  __syncthreads();
  if (t < 16) { float s = b2[t]; for (int k = 0; k < 32; k++) s += y1[k] * w2[k * 16 + t]; y2[t] = fmaxf(s, 0.f); }
  __syncthreads();
  if (t < 2) { float s = b3[t]; for (int k = 0; k < 16; k++) s += y2[k] * w3[k * 2 + t]; out[g * 2 + t] = s; }
}

extern "C" void kernel_launch(void* const* d_in, const int* in_sizes, int n_in,
                              void* d_out, int out_size, void* d_ws, size_t ws_size,
                              hipStream_t stream) {
  const float* h_in = (const float*)d_in[0];
  const float* e_in = (const float*)d_in[1];
  const int* src = (const int*)d_in[2];
  const int* dst = (const int*)d_in[3];
  const int* gid = (const int*)d_in[4];

  // params leaves in JAX pytree order (dict keys sorted):
  // emb_e.{b,w}, emb_h.{b,w}, layers[0..2].{M.b,M.w,U.b,U.w,bn_b,bn_g,mix.b,mix.w}, mlp[0..2].{b,w}
  static const int leafsz[34] = {
      64, 1024, 64, 2048,
      64, 12288, 64, 53248, 64, 64, 64, 4096,
      64, 12288, 64, 53248, 64, 64, 64, 4096,
      64, 12288, 64, 28672, 64, 64, 64, 4096,
      32, 2048, 16, 512, 2, 32};
  const float* P[34];
  if (n_in >= 39) {
    for (int i = 0; i < 34; i++) P[i] = (const float*)d_in[5 + i];
  } else {
    const float* base = (const float*)d_in[5];
    size_t off = 0;
    for (int i = 0; i < 34; i++) { P[i] = base + off; off += (size_t)leafsz[i]; }
  }

  // workspace carve (256B aligned)
  char* wp = (char*)d_ws;
  auto carve = [&](size_t bytes) -> char* {
    char* p = wp; wp += (bytes + 255) & ~(size_t)255; return p;
  };
  float* deg   = (float*)carve((size_t)NN * 4);
  float* hA    = (float*)carve((size_t)NN * 64 * 4);
  float* hB    = (float*)carve((size_t)NN * 64 * 4);
  float* sum   = (float*)carve((size_t)NN * 64 * 4);  // sumsq must follow contiguously
  float* sumsq = (float*)carve((size_t)NN * 64 * 4);
  unsigned* maxe = (unsigned*)carve((size_t)NN * 64 * 4);
  unsigned* mine = (unsigned*)carve((size_t)NN * 64 * 4);
  float* zbuf  = (float*)carve((size_t)NN * 64 * 4);
  float* bnsum = (float*)carve(64 * 4);  // bnss follows contiguously (256B each)
  float* bnss  = (float*)carve(64 * 4);
  float* alpha = (float*)carve(64 * 4);
  float* beta  = (float*)carve(64 * 4);
  float* wfold = (float*)carve(16 * 64 * 4);
  float* bfold = (float*)carve(64 * 4);
  float* hg    = (float*)carve((size_t)NG * 64 * 4);
  float* cnt   = (float*)carve((size_t)NG * 4);
  (void)ws_size; (void)in_sizes; (void)out_size;

  hipMemsetAsync(deg, 0, (size_t)NN * 4, stream);
  deg_kernel<<<(NE + 255) / 256, 256, 0, stream>>>(dst, deg, NE);

  int ntiles = (NN + 15) / 16, nblocks = (ntiles + 3) / 4;
  embed_kernel<<<nblocks, 128, 0, stream>>>(h_in, P[3], P[2], hA, NN);

  float* hcur = hA;
  float* hnext = hB;
  for (int l = 0; l < 3; l++) {
    int pb = 4 + l * 8;  // M.b, M.w, U.b, U.w, bn_b, bn_g, mix.b, mix.w
    fold_kernel<<<5, 256, 0, stream>>>(P[1], P[0], P[pb + 1], P[pb + 0], wfold, bfold);

    hipMemsetAsync(sum, 0, (size_t)NN * 64 * 4 * 2, stream);   // sum + sumsq
    hipMemsetAsync(maxe, 0x00, (size_t)NN * 64 * 4, stream);   // ordered-encoding minimum
    hipMemsetAsync(mine, 0xFF, (size_t)NN * 64 * 4, stream);   // ordered-encoding maximum

    int etiles = (NE + 15) / 16, eblocks = (etiles + 3) / 4;
    msg_agg_kernel<<<eblocks, 128, 0, stream>>>(hcur, e_in, src, dst, P[pb + 1], wfold, bfold,
                                                sum, sumsq, maxe, mine, NE, (l < 2) ? 1 : 0);

    u_kernel<<<nblocks, 128, 0, stream>>>(hcur, sum, sumsq, maxe, mine, deg,
                                          P[pb + 3], P[pb + 2], zbuf, NN, (l < 2) ? 0 : 1);

    hipMemsetAsync(bnsum, 0, 512, stream);  // bnsum + bnss
    bn_stats_kernel<<<256, 256, 0, stream>>>(zbuf, bnsum, bnss, NN);
    bn_affine_kernel<<<1, 64, 0, stream>>>(bnsum, bnss, P[pb + 5], P[pb + 4], alpha, beta, (float)NN);
    mix_kernel<<<nblocks, 128, 0, stream>>>(zbuf, alpha, beta, P[pb + 7], P[pb + 6], hcur, hnext, NN);

    float* tmp = hcur; hcur = hnext; hnext = tmp;
  }

  hipMemsetAsync(hg, 0, (size_t)NG * 64 * 4, stream);
  hipMemsetAsync(cnt, 0, (size_t)NG * 4, stream);
  pool_kernel<<<((size_t)NN * 64 + 255) / 256, 256, 0, stream>>>(hcur, gid, hg, cnt, NN);
  mlp_kernel<<<NG, 64, 0, stream>>>(hg, cnt, P[28], P[29], P[30], P[31], P[32], P[33], (float*)d_out);
}